// SMILE_47794396070562
// MI455X (gfx1250) — compile-verified
//
#include <hip/hip_runtime.h>
#include <hip/hip_bf16.h>
#include <math.h>

typedef float v2f __attribute__((ext_vector_type(2)));
typedef float v8f __attribute__((ext_vector_type(8)));

#define NROWS 1024
#define HDIM  128
#define CLIPV 10.0f

// ---------------------------------------------------------------------------
// Stage 0: zero the double accumulators (sumExp, diagSum) in workspace.
// ---------------------------------------------------------------------------
__global__ void smile_init(double* acc) {
    acc[0] = 0.0;
    acc[1] = 0.0;
}

// ---------------------------------------------------------------------------
// Stage 1: exact-fp32 WMMA GEMM (V_WMMA_F32_16X16X4_F32).
//   z==0: Aprime = z_c @ w1[0:128,:]   + b1   (b1 folded here)
//   z==1: Bmat   = z_d @ w1[128:256,:]
// One wave per 16x16 output tile; K looped in chunks of 4.
// ---------------------------------------------------------------------------
__global__ __launch_bounds__(32)
void smile_gemm(const float* __restrict__ z_c, const float* __restrict__ z_d,
                const float* __restrict__ w1,  const float* __restrict__ b1,
                float* __restrict__ Aprime,    float* __restrict__ Bmat) {
    const int z  = blockIdx.z;
    const float* X    = z ? z_d : z_c;
    const float* W    = w1 + z * HDIM * HDIM;      // row stride HDIM
    float*       Out  = z ? Bmat : Aprime;
    const bool   addB = (z == 0);

    const int m0   = blockIdx.x * 16;
    const int n0   = blockIdx.y * 16;
    const int lane = threadIdx.x & 31;
    const int lo   = lane & 15;
    const int hi   = lane >> 4;

    v8f c = {0.f, 0.f, 0.f, 0.f, 0.f, 0.f, 0.f, 0.f};

    const float* xrow = X + (m0 + lo) * HDIM;
    #pragma unroll 4
    for (int t = 0; t < HDIM / 4; ++t) {
        const int kc = 4 * t + 2 * hi;
        v2f a;
        a[0] = xrow[kc + 0];
        a[1] = xrow[kc + 1];
        v2f b;
        const float* wp = W + kc * HDIM + (n0 + lo);
        b[0] = wp[0];
        b[1] = wp[HDIM];
        c = __builtin_amdgcn_wmma_f32_16x16x4_f32(
                /*neg_a=*/false, a, /*neg_b=*/false, b,
                /*c_mod=*/(short)0, c, /*reuse_a=*/false, /*reuse_b=*/false);
    }

    const float bb = addB ? b1[n0 + lo] : 0.0f;
    #pragma unroll
    for (int r = 0; r < 8; ++r) {
        const int row = m0 + r + 8 * hi;
        Out[row * HDIM + (n0 + lo)] = c[r] + bb;
    }
}

// ---------------------------------------------------------------------------
// Stage 2: T1[i,j] = sum_k relu(Aprime[j,k] + Bmat[i,k]) * w2[k] + b2
//
// 32(i) x 64(j) tile per 128-thread block, 4x4 register tile per thread.
// LDS tiles stored TRANSPOSED ([k][j], [k][i]) with 16B-aligned padded row
// strides (68 / 36 floats) so the hot loop needs only two ds_load_b128 per k
// (conflict-free: byte addr = 272k + 16*tj spans all 64 banks).
// w2[k] is read with a uniform address -> scalar s_load (constant cache).
// Tile fill uses CDNA5 async-to-LDS (ASYNCcnt) copies.
// ---------------------------------------------------------------------------
#define TI 32
#define TJ 64
#define AST 68   // As_t row stride (floats): 68*4=272 B, 16B aligned, bank-friendly
#define BST 36   // Bs_t row stride (floats): 36*4=144 B

// One 4-byte async global->LDS copy (per-lane addresses). LDS offset is the
// low 32 bits of the generic pointer (== wave-relative LDS byte offset).
#define ASYNC_B32(ldsp, gp)                                                  \
    asm volatile("global_load_async_to_lds_b32 %0, %1, off"                  \
                 :: "v"((unsigned)(uintptr_t)(ldsp)),                        \
                    "v"((unsigned long long)(uintptr_t)(gp))                 \
                 : "memory")

__global__ __launch_bounds__(128)
void smile_pairs(const float* __restrict__ Aprime, const float* __restrict__ Bmat,
                 const float* __restrict__ w2, const float* __restrict__ b2p,
                 double* __restrict__ acc) {
    __shared__ float As_t[HDIM * AST];   // [k][j]  34,816 B
    __shared__ float Bs_t[HDIM * BST];   // [k][i]  18,432 B
    __shared__ double redE[128];
    __shared__ double redD[128];

    const int tid = threadIdx.x;
    const int i0 = blockIdx.y * TI;
    const int j0 = blockIdx.x * TJ;

    // --- async tile fill with transpose ---
    // lane = k (coalesced global reads), loop = row; LDS writes stride 272 B
    // (16 banks x 2-way) / 144 B.
    for (int j = 0; j < TJ; ++j)
        ASYNC_B32(&As_t[tid * AST + j], &Aprime[(j0 + j) * HDIM + tid]);
    for (int i = 0; i < TI; ++i)
        ASYNC_B32(&Bs_t[tid * BST + i], &Bmat[(i0 + i) * HDIM + tid]);
    asm volatile("s_wait_asynccnt 0x0" ::: "memory");
    __syncthreads();

    const int ti = tid >> 4;   // 0..7   -> 4 i rows each
    const int tj = tid & 15;   // 0..15  -> 4 j cols each

    float accr[4][4];
    #pragma unroll
    for (int ii = 0; ii < 4; ++ii)
        #pragma unroll
        for (int jj = 0; jj < 4; ++jj) accr[ii][jj] = 0.0f;

    #pragma unroll 4
    for (int k = 0; k < HDIM; ++k) {
        const float4 av = *reinterpret_cast<const float4*>(&As_t[k * AST + 4 * tj]);
        const float4 bv = *reinterpret_cast<const float4*>(&Bs_t[k * BST + 4 * ti]);
        const float w2k = w2[k];                    // uniform -> s_load
        const float aj[4] = {av.x, av.y, av.z, av.w};
        const float bi[4] = {bv.x, bv.y, bv.z, bv.w};
        #pragma unroll
        for (int ii = 0; ii < 4; ++ii)
            #pragma unroll
            for (int jj = 0; jj < 4; ++jj) {
                float t = aj[jj] + bi[ii];
                t = fmaxf(t, 0.0f);
                accr[ii][jj] = fmaf(t, w2k, accr[ii][jj]);
            }
    }

    // --- epilogue: diag sum + logsumexp partials ---
    const float b2v = b2p[0];
    double sE = 0.0, sD = 0.0;
    #pragma unroll
    for (int ii = 0; ii < 4; ++ii) {
        const int gi = i0 + 4 * ti + ii;
        #pragma unroll
        for (int jj = 0; jj < 4; ++jj) {
            const int gj = j0 + 4 * tj + jj;
            const float t1 = accr[ii][jj] + b2v;
            if (gi == gj) {
                sD += (double)t1;                       // T0 uses unclipped T1
            } else {
                const float cc = fminf(fmaxf(t1, -CLIPV), CLIPV);
                sE += (double)__expf(cc - CLIPV);       // shift by clip max
            }
        }
    }

    redE[tid] = sE;
    redD[tid] = sD;
    __syncthreads();
    for (int s = 64; s > 0; s >>= 1) {
        if (tid < s) { redE[tid] += redE[tid + s]; redD[tid] += redD[tid + s]; }
        __syncthreads();
    }
    if (tid == 0) {
        atomicAdd(&acc[0], redE[0]);   // global_atomic_add_f64
        atomicAdd(&acc[1], redD[0]);
    }
}

// ---------------------------------------------------------------------------
// Stage 3: dv = mean(diag) - (log(sumExp) + CLIP - log(N*(N-1)))
// ---------------------------------------------------------------------------
__global__ void smile_final(const double* __restrict__ acc, float* __restrict__ out) {
    const double T0  = acc[1] / (double)NROWS;
    const double lme = log(acc[0]) + (double)CLIPV
                     - log((double)NROWS * (double)(NROWS - 1));
    out[0] = (float)(T0 - lme);
}

// ---------------------------------------------------------------------------
extern "C" void kernel_launch(void* const* d_in, const int* in_sizes, int n_in,
                              void* d_out, int out_size, void* d_ws, size_t ws_size,
                              hipStream_t stream) {
    (void)in_sizes; (void)n_in; (void)out_size; (void)ws_size;
    const float* z_c = (const float*)d_in[0];
    const float* z_d = (const float*)d_in[1];
    const float* w1  = (const float*)d_in[2];
    const float* b1  = (const float*)d_in[3];
    const float* w2  = (const float*)d_in[4];
    const float* b2  = (const float*)d_in[5];

    char*   ws   = (char*)d_ws;
    double* acc  = (double*)ws;                                  // 2 doubles
    float*  Ap   = (float*)(ws + 256);                           // 1024x128 f32
    float*  Bm   = (float*)(ws + 256 + NROWS * HDIM * sizeof(float));

    smile_init<<<1, 1, 0, stream>>>(acc);

    smile_gemm<<<dim3(NROWS / 16, HDIM / 16, 2), 32, 0, stream>>>(
        z_c, z_d, w1, b1, Ap, Bm);

    smile_pairs<<<dim3(NROWS / TJ, NROWS / TI), 128, 0, stream>>>(
        Ap, Bm, w2, b2, acc);

    smile_final<<<1, 1, 0, stream>>>(acc, (float*)d_out);
}